// BoundarySuppressionWithSmoothing_19250043420931
// MI455X (gfx1250) — compile-verified
//
#include <hip/hip_runtime.h>

typedef float v2f __attribute__((ext_vector_type(2)));
typedef float v8f __attribute__((ext_vector_type(8)));

#define IMG_H 512
#define IMG_W 1024
#define IMG_B 8
#define TS 32

// ---------------------------------------------------------------------------
// CDNA5 async global->LDS path (ASYNCcnt-tracked), guarded so the fallback
// keeps the build green if the toolchain lacks the gfx1250 builtins.
// Parameter types per hipcc diagnostics: b32 takes global int*, b128 takes
// global int __vector(4)*.
// ---------------------------------------------------------------------------
#if defined(__has_builtin)
#if __has_builtin(__builtin_amdgcn_global_load_async_to_lds_b128) && \
    __has_builtin(__builtin_amdgcn_global_load_async_to_lds_b32) &&  \
    __has_builtin(__builtin_amdgcn_s_wait_asynccnt)
#define HAVE_ASYNC_LDS 1
#endif
#endif
#ifndef HAVE_ASYNC_LDS
#define HAVE_ASYNC_LDS 0
#endif

#if HAVE_ASYNC_LDS
typedef int v4i_b __attribute__((vector_size(16)));

__device__ __forceinline__ __attribute__((address_space(1))) int* gptr_i32(const void* p) {
  return (__attribute__((address_space(1))) int*)(uintptr_t)p;       // same 64-bit VA
}
__device__ __forceinline__ __attribute__((address_space(3))) int* lptr_i32(void* p) {
  // generic LDS pointers keep the LDS byte offset in addr[31:0] (aperture in [63:32])
  return (__attribute__((address_space(3))) int*)(uint32_t)(uintptr_t)p;
}
__device__ __forceinline__ __attribute__((address_space(1))) v4i_b* gptr_v4(const void* p) {
  return (__attribute__((address_space(1))) v4i_b*)(uintptr_t)p;
}
__device__ __forceinline__ __attribute__((address_space(3))) v4i_b* lptr_v4(void* p) {
  return (__attribute__((address_space(3))) v4i_b*)(uint32_t)(uintptr_t)p;
}
#endif

// normalized 1-D Gaussian (sigma=1, 7 taps); banded-matrix weight:
// non-zero only at band offsets {0,6,12,18,24,30,36}. Branch-free select chain,
// inline constants only -> no division, no memory, no EXEC divergence.
__device__ __forceinline__ float bandw(int e) {
  float w = 0.0f;
  w = (e == 0)  ? 0.00443305f : w;
  w = (e == 6)  ? 0.05400558f : w;
  w = (e == 12) ? 0.24203623f : w;
  w = (e == 18) ? 0.39905028f : w;
  w = (e == 24) ? 0.24203623f : w;
  w = (e == 30) ? 0.05400558f : w;
  w = (e == 36) ? 0.00443305f : w;
  return w;
}

// ---------------------------------------------------------------------------
// Kernel 1: fused boundary detection + 4 masked-averaging iterations.
// One 32x32 tile per 256-thread block, everything staged in LDS.
// ---------------------------------------------------------------------------
__device__ __forceinline__ void dilate_cross(const float* src, int sN, int sH,
                                             float* dst, int dN, int dH,
                                             int ty0, int tx0, int tid) {
  for (int i = tid; i < dN * dN; i += 256) {
    int ly = i / dN, lx = i % dN;
    int gy = ty0 - dH + ly, gx = tx0 - dH + lx;
    if (gy < 0 || gy >= IMG_H || gx < 0 || gx >= IMG_W) continue;
    int cy = gy - ty0 + sH, cx = gx - tx0 + sH;
    float m = src[cy * sN + cx];
    if (gy > 0)         m = fmaxf(m, src[(cy - 1) * sN + cx]);
    if (gy < IMG_H - 1) m = fmaxf(m, src[(cy + 1) * sN + cx]);
    if (gx > 0)         m = fmaxf(m, src[cy * sN + cx - 1]);
    if (gx < IMG_W - 1) m = fmaxf(m, src[cy * sN + cx + 1]);
    dst[i] = m;
  }
}

__device__ __forceinline__ void smooth_step(const float* cur, float* nxt,
                                            const float* eb, int eN, int eH,
                                            int h, int ty0, int tx0, int tid) {
  int n = TS + 2 * h;
  for (int i = tid; i < n * n; i += 256) {
    int ly = i / n, lx = i % n;
    int gy = ty0 - h + ly, gx = tx0 - h + lx;
    if (gy < 0 || gy >= IMG_H || gx < 0 || gx >= IMG_W) continue;
    float eb0 = eb[(gy - ty0 + eH) * eN + (gx - tx0 + eH)];
    float pv = cur[(gy - ty0 + 4) * 40 + (gx - tx0 + 4)];
    float nv = pv;
    if (eb0 != 0.0f) {                 // mask==0 -> replace by masked 3x3 average
      float acc = 0.0f; int cnt = 0;
      #pragma unroll
      for (int dy = -1; dy <= 1; ++dy) {
        #pragma unroll
        for (int dx = -1; dx <= 1; ++dx) {
          int ny = min(max(gy + dy, 0), IMG_H - 1);   // 'edge' padding
          int nx = min(max(gx + dx, 0), IMG_W - 1);
          float ebn = eb[(ny - ty0 + eH) * eN + (nx - tx0 + eH)];
          if (ebn == 0.0f) {
            acc += cur[(ny - ty0 + 4) * 40 + (nx - tx0 + 4)];
            cnt++;
          }
        }
      }
      if (cnt != 0) nv = acc / (float)cnt;
    }
    nxt[(gy - ty0 + 4) * 40 + (gx - tx0 + 4)] = nv;
  }
}

__global__ __launch_bounds__(256) void suppress_kernel(const float* __restrict__ x,
                                                       const int* __restrict__ pred,
                                                       float* __restrict__ out) {
  __shared__ float lab[48 * 48];   // halo 8
  __shared__ float bnd[46 * 46];   // halo 7
  __shared__ float d1[44 * 44];    // halo 6
  __shared__ float d2[42 * 42];    // halo 5
  __shared__ float d3[40 * 40];    // halo 4
  __shared__ float bufA[40 * 40];  // halo 4
  __shared__ float bufB[40 * 40];

  const int tx0 = blockIdx.x * TS;
  const int ty0 = blockIdx.y * TS;
  const long base = (long)blockIdx.z * IMG_H * IMG_W;
  const int tid = threadIdx.x;

  for (int i = tid; i < 48 * 48; i += 256) {
    int ly = i / 48, lx = i % 48;
    int gy = ty0 - 8 + ly, gx = tx0 - 8 + lx;
    if (gy >= 0 && gy < IMG_H && gx >= 0 && gx < IMG_W)
      lab[i] = (float)pred[base + (long)gy * IMG_W + gx];
  }
  for (int i = tid; i < 40 * 40; i += 256) {
    int ly = i / 40, lx = i % 40;
    int gy = ty0 - 4 + ly, gx = tx0 - 4 + lx;
    if (gy >= 0 && gy < IMG_H && gx >= 0 && gx < IMG_W)
      bufA[i] = x[base + (long)gy * IMG_W + gx];
  }
  __syncthreads();

  // boundaries = (cross_dilate(lab) != erode3(lab)); OOB neighbors ignored
  for (int i = tid; i < 46 * 46; i += 256) {
    int ly = i / 46, lx = i % 46;
    int gy = ty0 - 7 + ly, gx = tx0 - 7 + lx;
    if (gy < 0 || gy >= IMG_H || gx < 0 || gx >= IMG_W) continue;
    int cy = ly + 1, cx = lx + 1;
    float c = lab[cy * 48 + cx];
    float mx = c, mn = c;
    if (gy > 0)          mx = fmaxf(mx, lab[(cy - 1) * 48 + cx]);
    if (gy < IMG_H - 1)  mx = fmaxf(mx, lab[(cy + 1) * 48 + cx]);
    if (gx > 0)          mx = fmaxf(mx, lab[cy * 48 + cx - 1]);
    if (gx < IMG_W - 1)  mx = fmaxf(mx, lab[cy * 48 + cx + 1]);
    #pragma unroll
    for (int dy = -1; dy <= 1; ++dy) {
      #pragma unroll
      for (int dx = -1; dx <= 1; ++dx) {
        int ny = gy + dy, nx = gx + dx;
        if (ny < 0 || ny >= IMG_H || nx < 0 || nx >= IMG_W) continue;
        mn = fminf(mn, lab[(cy + dy) * 48 + cx + dx]);
      }
    }
    bnd[i] = (mx != mn) ? 1.0f : 0.0f;
  }
  __syncthreads();

  dilate_cross(bnd, 46, 7, d1, 44, 6, ty0, tx0, tid); __syncthreads();
  dilate_cross(d1, 44, 6, d2, 42, 5, ty0, tx0, tid);  __syncthreads();
  dilate_cross(d2, 42, 5, d3, 40, 4, ty0, tx0, tid);  __syncthreads();

  // 4 iterations: it uses eb dilated r = 3-it times, writes halo 3-it
  smooth_step(bufA, bufB, d3, 40, 4, 3, ty0, tx0, tid); __syncthreads();
  smooth_step(bufB, bufA, d2, 42, 5, 2, ty0, tx0, tid); __syncthreads();
  smooth_step(bufA, bufB, d1, 44, 6, 1, ty0, tx0, tid); __syncthreads();
  smooth_step(bufB, bufA, bnd, 46, 7, 0, ty0, tx0, tid); __syncthreads();

  for (int i = tid; i < TS * TS; i += 256) {
    int ly = i / TS, lx = i % TS;
    out[base + (long)(ty0 + ly) * IMG_W + (tx0 + lx)] = bufA[(ly + 4) * 40 + (lx + 4)];
  }
}

// ---------------------------------------------------------------------------
// Kernel 2: horizontal 7-tap dilation-6 Gaussian as banded WMMA mat-mul.
// out(16x16) = A(data 16x52) x B(banded weights 52x16), 13x K=4 chunks.
// Staging uses async global->LDS (b32 per element: column clamp breaks b128).
// ---------------------------------------------------------------------------
__global__ __launch_bounds__(256) void gauss_h_kernel(const float* __restrict__ in,
                                                      float* __restrict__ out) {
  __shared__ float s[16 * 168];           // 16 rows x 164 staged cols (stride 168)
  const int x0 = blockIdx.x * 128;
  const int y0 = blockIdx.y * 16;         // flat row over B*H = 4096
  const int tid = threadIdx.x;

#if HAVE_ASYNC_LDS
  for (int i = tid; i < 16 * 164; i += 256) {
    int ly = i / 164, lx = i % 164;
    int gc = min(max(x0 - 18 + lx, 0), IMG_W - 1);   // 'edge' padding
    __builtin_amdgcn_global_load_async_to_lds_b32(
        gptr_i32(in + (long)(y0 + ly) * IMG_W + gc),
        lptr_i32(s + ly * 168 + lx), 0, 0);
  }
  __builtin_amdgcn_s_wait_asynccnt(0);
#else
  for (int i = tid; i < 16 * 164; i += 256) {
    int ly = i / 164, lx = i % 164;
    int gc = min(max(x0 - 18 + lx, 0), IMG_W - 1);
    s[ly * 168 + lx] = in[(long)(y0 + ly) * IMG_W + gc];
  }
#endif
  __syncthreads();

  const int wave = tid >> 5;
  const int lane = tid & 31;
  const int m = lane & 15;            // A row / B col / D col
  const int koff = (lane >> 4) * 2;   // K sub-index within a 4-chunk
  const int arow = m * 168 + 16 * wave;

  v8f acc = {};
  #pragma unroll
  for (int c = 0; c < 13; ++c) {
    int k0 = 4 * c + koff;
    v2f a, b;
    a.x = s[arow + k0];
    a.y = s[arow + k0 + 1];
    b.x = bandw(k0 - m);
    b.y = bandw(k0 + 1 - m);
    acc = __builtin_amdgcn_wmma_f32_16x16x4_f32(false, a, false, b,
                                                (short)0, acc, false, false);
  }

  const int col = x0 + 16 * wave + m;
  const int rsel = (lane >> 4) * 8;
  #pragma unroll
  for (int v = 0; v < 8; ++v)
    out[(long)(y0 + v + rsel) * IMG_W + col] = acc[v];
}

// ---------------------------------------------------------------------------
// Kernel 3: vertical pass: out(16x16) = A(banded weights 16x52) x B(data 52x16).
// Row-granular clamp keeps rows contiguous -> async b128 staging (512B/wave-op).
// ---------------------------------------------------------------------------
__global__ __launch_bounds__(256) void gauss_v_kernel(const float* __restrict__ in,
                                                      float* __restrict__ out) {
  __shared__ float s[52 * 128];
  const int x0 = blockIdx.x * 128;
  const int b = blockIdx.y >> 5;          // 32 row-tiles per image
  const int y0 = (blockIdx.y & 31) * 16;
  const long base = (long)b * IMG_H * IMG_W;
  const int tid = threadIdx.x;
  const int wave = tid >> 5;
  const int lane = tid & 31;

#if HAVE_ASYNC_LDS
  // each wave-op moves one 128-float row: lane carries 16B
  for (int k = wave; k < 52; k += 8) {
    int gy = min(max(y0 - 18 + k, 0), IMG_H - 1);    // 'edge' padding inside image
    __builtin_amdgcn_global_load_async_to_lds_b128(
        gptr_v4(in + base + (long)gy * IMG_W + x0 + lane * 4),
        lptr_v4(s + k * 128 + lane * 4), 0, 0);
  }
  __builtin_amdgcn_s_wait_asynccnt(0);
#else
  for (int i = tid; i < 52 * 128; i += 256) {
    int k = i >> 7, lx = i & 127;
    int gy = min(max(y0 - 18 + k, 0), IMG_H - 1);
    s[i] = in[base + (long)gy * IMG_W + (x0 + lx)];
  }
#endif
  __syncthreads();

  const int m = lane & 15;
  const int koff = (lane >> 4) * 2;
  const int bcol = 16 * wave + m;

  v8f acc = {};
  #pragma unroll
  for (int c = 0; c < 13; ++c) {
    int k0 = 4 * c + koff;
    v2f a, bfrag;
    a.x = bandw(k0 - m);
    a.y = bandw(k0 + 1 - m);
    bfrag.x = s[k0 * 128 + bcol];
    bfrag.y = s[(k0 + 1) * 128 + bcol];
    acc = __builtin_amdgcn_wmma_f32_16x16x4_f32(false, a, false, bfrag,
                                                (short)0, acc, false, false);
  }

  const int col = x0 + bcol;
  const int rsel = (lane >> 4) * 8;
  #pragma unroll
  for (int v = 0; v < 8; ++v)
    out[base + (long)(y0 + v + rsel) * IMG_W + col] = acc[v];
}

// ---------------------------------------------------------------------------
extern "C" void kernel_launch(void* const* d_in, const int* in_sizes, int n_in,
                              void* d_out, int out_size, void* d_ws, size_t ws_size,
                              hipStream_t stream) {
  (void)in_sizes; (void)n_in; (void)out_size; (void)ws_size;
  const float* x = (const float*)d_in[0];
  const int* pred = (const int*)d_in[1];
  float* out = (float*)d_out;
  const size_t N = (size_t)IMG_B * IMG_H * IMG_W;
  float* ws0 = (float*)d_ws;
  float* ws1 = ws0 + N;

  suppress_kernel<<<dim3(IMG_W / TS, IMG_H / TS, IMG_B), 256, 0, stream>>>(x, pred, ws0);
  gauss_h_kernel<<<dim3(IMG_W / 128, (IMG_B * IMG_H) / 16), 256, 0, stream>>>(ws0, ws1);
  gauss_v_kernel<<<dim3(IMG_W / 128, IMG_B * (IMG_H / 16)), 256, 0, stream>>>(ws1, out);
}